// GCN_20701742367344
// MI455X (gfx1250) — compile-verified
//
#include <hip/hip_runtime.h>
#include <hip/hip_bf16.h>

typedef __attribute__((ext_vector_type(2))) float v2f;
typedef __attribute__((ext_vector_type(8))) float v8f;

#define HID 128

// ---------------------------------------------------------------------------
// Build xpad[N,132] = concat(h[N,128], coords[N,3], 0-pad)
// ---------------------------------------------------------------------------
__global__ void build_x0_k(const float* __restrict__ h,
                           const float* __restrict__ coords,
                           float* __restrict__ xp, int n) {
  size_t total = (size_t)n * 132;
  for (size_t i = (size_t)blockIdx.x * blockDim.x + threadIdx.x; i < total;
       i += (size_t)gridDim.x * blockDim.x) {
    size_t node = i / 132;
    int col = (int)(i - node * 132);
    float v;
    if (col < 128)       v = h[node * 128 + col];
    else if (col < 131)  v = coords[node * 3 + (col - 128)];
    else                 v = 0.0f;
    xp[i] = v;
  }
}

// ---------------------------------------------------------------------------
// Pad W0[131,128] -> Wp[132,128] (last row zero) so the GEMM is branchless.
// ---------------------------------------------------------------------------
__global__ void padW_k(const float* __restrict__ W, float* __restrict__ Wp) {
  int i = blockIdx.x * blockDim.x + threadIdx.x;
  if (i < 132 * HID) Wp[i] = (i < 131 * HID) ? W[i] : 0.0f;
}

// ---------------------------------------------------------------------------
// Degree: deg = 1 (self loop) + count of incoming edges; dinv = rsqrt(deg)
// ---------------------------------------------------------------------------
__global__ void deg_init_k(float* __restrict__ deg, int n) {
  int i = blockIdx.x * blockDim.x + threadIdx.x;
  if (i < n) deg[i] = 1.0f;
}

__global__ void deg_count_k(const long long* __restrict__ dst,
                            float* __restrict__ deg, long long E) {
  long long e = (long long)blockIdx.x * blockDim.x + threadIdx.x;
  if (e < E) atomicAdd(&deg[(int)dst[e]], 1.0f);
}

__global__ void dinv_k(const float* __restrict__ deg,
                       float* __restrict__ dinv, int n) {
  int i = blockIdx.x * blockDim.x + threadIdx.x;
  if (i < n) dinv[i] = rsqrtf(deg[i]);
}

// ---------------------------------------------------------------------------
// Y[n,128] = X[n,KPAD] @ W[KPAD,128] via V_WMMA_F32_16X16X4_F32.
// Block = 256 threads = 8 waves = the 8 column tiles of one 16-row stripe.
// The 16xKPAD X stripe is staged once in LDS (coalesced) and shared by all
// 8 waves; A-frags come from LDS (8B-aligned ds_load, conflict-free for KPAD=132: lane
// bank = 4*lane mod 64; KPAD=128 maps all 16 rows to one bank). K loop fully unrolls -> KPAD/4 back-to-back wmmas.
// Fragment layouts per CDNA5 ISA 7.12.2:
//   A (16x4 f32): lanes 0-15 row=lane K={k4,k4+1}; lanes 16-31 K={k4+2,k4+3}
//   B (4x16 f32): mirrored (half selects K pair, lane%16 = N)
//   C/D (16x16):  vgpr v, half h -> row v+8h, col lane%16
// ---------------------------------------------------------------------------
template <int KPAD>
__global__ void __launch_bounds__(256)
gemm_wmma_k(const float* __restrict__ X, const float* __restrict__ W,
            float* __restrict__ Y) {
  __shared__ float xs[16 * KPAD];
  const int tid  = threadIdx.x;
  const int lane = tid & 31;
  const int wid  = tid >> 5;
  const int mrow = blockIdx.x * 16;

  // Stage the 16-row X stripe (coalesced: consecutive tid -> consecutive col).
  for (int i = tid; i < 16 * KPAD; i += 256)
    xs[i] = X[(size_t)mrow * KPAD + i];
  __syncthreads();

  const int ncol = wid * 16;       // this wave's 16 output columns
  const int half = lane >> 4;      // 0 or 1
  const int lr   = lane & 15;

  v8f c = {};
#pragma unroll
  for (int k4 = 0; k4 < KPAD; k4 += 4) {
    const int ka = k4 + 2 * half;                    // even -> 8B aligned
    v2f a = *(const v2f*)&xs[lr * KPAD + ka];
    v2f b;
    b.x = W[(size_t)ka * HID + ncol + lr];           // coalesced per half-wave
    b.y = W[(size_t)(ka + 1) * HID + ncol + lr];
    c = __builtin_amdgcn_wmma_f32_16x16x4_f32(
        /*neg_a=*/false, a, /*neg_b=*/false, b,
        /*c_mod=*/(short)0, c, /*reuse_a=*/false, /*reuse_b=*/false);
  }

#pragma unroll
  for (int v = 0; v < 8; ++v)
    Y[(size_t)(mrow + v + 8 * half) * HID + ncol + lr] = c[v];
}

// ---------------------------------------------------------------------------
// agg[i,k] = y[i,k] * dinv[i]^2 + bias[k]   (self-loop term + bias, inits agg)
// ---------------------------------------------------------------------------
__global__ void agg_init_k(const float* __restrict__ y,
                           const float* __restrict__ dinv,
                           const float* __restrict__ bias,
                           float* __restrict__ agg, int n) {
  int node = blockIdx.x;
  int k = threadIdx.x;
  float di = dinv[node];
  size_t idx = (size_t)node * HID + k;
  agg[idx] = y[idx] * di * di + bias[k];
}

// ---------------------------------------------------------------------------
// Edge scatter: agg[dst] += y[src] * dinv[src]*dinv[dst]
// 32 lanes per edge, float4 per lane; atomics stay L2-resident (<<192MB).
// ---------------------------------------------------------------------------
__global__ void agg_scatter_k(const float* __restrict__ y,
                              const long long* __restrict__ src,
                              const long long* __restrict__ dst,
                              const float* __restrict__ dinv,
                              float* __restrict__ agg, long long E) {
  long long t = (long long)blockIdx.x * blockDim.x + threadIdx.x;
  long long e = t >> 5;
  int lane = (int)(t & 31);
  if (e >= E) return;
  int s = (int)src[e];
  int d = (int)dst[e];
  float coef = dinv[s] * dinv[d];
  float4 v = ((const float4*)(y + (size_t)s * HID))[lane];
  float* ad = agg + (size_t)d * HID + lane * 4;
  atomicAdd(ad + 0, v.x * coef);
  atomicAdd(ad + 1, v.y * coef);
  atomicAdd(ad + 2, v.z * coef);
  atomicAdd(ad + 3, v.w * coef);
}

// ---------------------------------------------------------------------------
// out = relu(agg)
// ---------------------------------------------------------------------------
__global__ void relu_k(const float* __restrict__ agg,
                       float* __restrict__ out, size_t total) {
  size_t i = (size_t)blockIdx.x * blockDim.x + threadIdx.x;
  if (i < total) out[i] = fmaxf(agg[i], 0.0f);
}

// ---------------------------------------------------------------------------
extern "C" void kernel_launch(void* const* d_in, const int* in_sizes, int n_in,
                              void* d_out, int out_size, void* d_ws, size_t ws_size,
                              hipStream_t stream) {
  const float* h      = (const float*)d_in[0];
  const float* coords = (const float*)d_in[1];
  const long long* edges = (const long long*)d_in[2];   // int64 in reference
  const float* Wts[4] = {(const float*)d_in[3], (const float*)d_in[5],
                         (const float*)d_in[7], (const float*)d_in[9]};
  const float* Bs[4]  = {(const float*)d_in[4], (const float*)d_in[6],
                         (const float*)d_in[8], (const float*)d_in[10]};

  const int n = in_sizes[0] / HID;            // 50000 (multiple of 16)
  const long long E = in_sizes[2] / 2;        // 1.6M
  const long long* src = edges;
  const long long* dst = edges + E;

  // Workspace (floats): xpad[n*132] (reused as 128-stride x for layers 1..3),
  // y[n*128], agg[n*128], deg[n], dinv[n], Wpad[132*128]  => ~78 MB.
  float* xpad = (float*)d_ws;
  float* y    = xpad + (size_t)n * 132;
  float* agg  = y    + (size_t)n * HID;
  float* deg  = agg  + (size_t)n * HID;
  float* dinv = deg  + n;
  float* Wpad = dinv + n;

  // ---- prep ---------------------------------------------------------------
  build_x0_k<<<4096, 256, 0, stream>>>(h, coords, xpad, n);
  padW_k<<<(132 * HID + 255) / 256, 256, 0, stream>>>(Wts[0], Wpad);
  deg_init_k<<<(n + 255) / 256, 256, 0, stream>>>(deg, n);
  deg_count_k<<<(int)((E + 255) / 256), 256, 0, stream>>>(dst, deg, E);
  dinv_k<<<(n + 255) / 256, 256, 0, stream>>>(deg, dinv, n);

  // ---- 4 GCN layers -------------------------------------------------------
  const size_t total = (size_t)n * HID;
  const int gemm_blocks = n / 16;             // one 16-row stripe per block
  const int scat_blocks = (int)((E * 32 + 255) / 256);

  for (int layer = 0; layer < 4; ++layer) {
    if (layer == 0)
      gemm_wmma_k<132><<<gemm_blocks, 256, 0, stream>>>(xpad, Wpad, y);
    else
      gemm_wmma_k<128><<<gemm_blocks, 256, 0, stream>>>(xpad, Wts[layer], y);
    agg_init_k<<<n, HID, 0, stream>>>(y, dinv, Bs[layer], agg, n);
    agg_scatter_k<<<scat_blocks, 256, 0, stream>>>(y, src, dst, dinv, agg, E);
    float* dest = (layer == 3) ? (float*)d_out : xpad;  // reuse xpad as next x
    relu_k<<<(int)((total + 255) / 256), 256, 0, stream>>>(agg, dest, total);
  }
}